// Final_stroke_finding_14053132993280
// MI455X (gfx1250) — compile-verified
//
#include <hip/hip_runtime.h>
#include <hip/hip_bf16.h>
#include <stdint.h>

// CDNA5 / gfx1250, wave32. D = 32 features everywhere.
typedef __attribute__((ext_vector_type(2))) float v2f;
typedef __attribute__((ext_vector_type(8))) float v8f;

__device__ __forceinline__ v8f wmma4(v2f a, v2f b, v8f c) {
  // V_WMMA_F32_16X16X4_F32 : D = A(16x4) * B(4x16) + C(16x16), fp32 throughout.
  return __builtin_amdgcn_wmma_f32_16x16x4_f32(false, a, false, b, (short)0, c,
                                               false, false);
}

// ---------------------------------------------------------------- elementwise
__global__ void k_xs(const float* __restrict__ x, const float* __restrict__ w,
                     float* __restrict__ xs, float* __restrict__ acc, int n) {
  int i = blockIdx.x * blockDim.x + threadIdx.x;
  if (i < n) {
    float v = x[i] * w[i >> 5];   // stroke_weights broadcast over 32 features
    xs[i] = v;
    acc[i] = v;                   // residual term of stroke_out
  }
}

__global__ void k_fill(uint32_t* __restrict__ p, uint32_t v, int n) {
  int i = blockIdx.x * blockDim.x + threadIdx.x;
  if (i < n) p[i] = v;
}

__global__ void k_acc_mean(float* __restrict__ acc, const float* __restrict__ s,
                           const float* __restrict__ c, int n) {
  int i = blockIdx.x * blockDim.x + threadIdx.x;
  if (i < n) acc[i] += s[i] / fmaxf(c[i >> 5], 1.0f);
}

__global__ void k_acc_sum(float* __restrict__ acc, const float* __restrict__ s, int n) {
  int i = blockIdx.x * blockDim.x + threadIdx.x;
  if (i < n) acc[i] += s[i];
}

__global__ void k_acc_max(float* __restrict__ acc, const float* __restrict__ m, int n) {
  int i = blockIdx.x * blockDim.x + threadIdx.x;
  if (i < n) {
    float v = m[i];
    // segment_max of an empty segment is -inf -> reference maps it to 0
    acc[i] += (__float_as_uint(v) == 0xFF800000u) ? 0.0f : v;
  }
}

// ------------------------------------------------------- [rows,32]@[32,32]+b
// One wave per 16-row tile; 2 N-tiles x 8 K-steps = 16 v_wmma per wave.
// A-row addresses are CLAMPED (not predicated): an A row only feeds the same
// D row, and stores are guarded only on the single ragged tail tile.
__global__ void __launch_bounds__(256)
k_gemm32(const float* __restrict__ in, const float* __restrict__ W,
         const float* __restrict__ bias, float* __restrict__ out, int nrows) {
  const int wave = threadIdx.x >> 5;
  const int lane = threadIdx.x & 31;
  const int half = lane >> 4;      // lanes 16-31 carry K+2/K+3 (A) and rows M+8 (D)
  const int lrow = lane & 15;
  const int tile = (blockIdx.x * 8 + wave) * 16;
  if (tile >= nrows) return;       // wave-uniform exit

  int row = tile + lrow;
  if (row >= nrows) row = nrows - 1;              // clamp, never zero
  const float* __restrict__ arow = in + row * 32 + half * 2;

  v8f c0 = {0, 0, 0, 0, 0, 0, 0, 0};
  v8f c1 = {0, 0, 0, 0, 0, 0, 0, 0};
#pragma unroll
  for (int k0 = 0; k0 < 32; k0 += 4) {
    const int ka = k0 + half * 2;
    v2f a = *(const v2f*)(arow + k0);             // one global_load_b64
    v2f b0, b1;
    b0.x = W[ka * 32 + lrow];
    b0.y = W[(ka + 1) * 32 + lrow];
    b1.x = W[ka * 32 + 16 + lrow];
    b1.y = W[(ka + 1) * 32 + 16 + lrow];
    c0 = wmma4(a, b0, c0);
    c1 = wmma4(a, b1, c1);
  }
  const float bb0 = bias[lrow];
  const float bb1 = bias[16 + lrow];
  if (tile + 16 <= nrows) {
    // full tile: unpredicated, branch-free epilogue (the common case)
#pragma unroll
    for (int r = 0; r < 8; ++r) {
      const int m = tile + r + half * 8;
      out[m * 32 + lrow]      = c0[r] + bb0;
      out[m * 32 + 16 + lrow] = c1[r] + bb1;
    }
  } else {
    // single ragged tail tile
#pragma unroll
    for (int r = 0; r < 8; ++r) {
      const int m = tile + r + half * 8;
      if (m < nrows) {
        out[m * 32 + lrow]      = c0[r] + bb0;
        out[m * 32 + 16 + lrow] = c1[r] + bb1;
      }
    }
  }
}

// ------------------------------------------------------------ edge scatters
// One wave32 per edge: edge id forced wave-uniform (readfirstlane) so the
// src/dst index fetches go down the scalar path; the 32 lanes then move the
// 32-float feature row as one coalesced 128B gather + one 128B atomic burst
// that resolves in the 192MB L2 (all accumulators are ~13MB).
__global__ void k_scatter_add(const float* __restrict__ t, const int* __restrict__ src,
                              const int* __restrict__ dst, float* __restrict__ s,
                              float* __restrict__ cnt, int nE) {
  int g = blockIdx.x * blockDim.x + threadIdx.x;
  int e = __builtin_amdgcn_readfirstlane(g >> 5);  // wave-uniform edge id
  int lane = threadIdx.x & 31;
  if (e >= nE) return;
  int si = src[e], di = dst[e];
  atomicAdd(&s[di * 32 + lane], t[si * 32 + lane]);
  if (cnt != nullptr && lane == 0) atomicAdd(&cnt[di], 1.0f);
}

__device__ __forceinline__ void atomicMaxF32(float* addr, float val) {
  // Sign-aware monotonic-bits trick; accumulator initialized to -inf bits.
  if (val >= 0.0f) atomicMax((int*)addr, __float_as_int(val));
  else             atomicMin((unsigned int*)addr, __float_as_uint(val));
}

__global__ void k_scatter_max(const float* __restrict__ t, const int* __restrict__ src,
                              const int* __restrict__ dst, float* __restrict__ s, int nE) {
  int g = blockIdx.x * blockDim.x + threadIdx.x;
  int e = __builtin_amdgcn_readfirstlane(g >> 5);
  int lane = threadIdx.x & 31;
  if (e >= nE) return;
  int si = src[e], di = dst[e];
  atomicMaxF32(&s[di * 32 + lane], t[si * 32 + lane]);
}

// --------------------------------------------------------------- fused decoder
// Per block: 64 nodes, 4 waves. relu residual -> LDS; feats=so@Wh+bh (WMMA);
// h=relu(fe@W1+b1) (WMMA); out=sigmoid(h@W2+b2) via per-node dot + shfl.
__global__ void __launch_bounds__(128)
k_decoder(const float* __restrict__ acc,
          const float* __restrict__ Wh, const float* __restrict__ bh,
          const float* __restrict__ W1, const float* __restrict__ b1,
          const float* __restrict__ W2, const float* __restrict__ b2,
          float* __restrict__ out, int n) {
  __shared__ float so[64 * 32];    //  8 KB
  __shared__ float fe[64 * 64];    // 16 KB
  __shared__ float hh[64 * 128];   // 32 KB
  const int tid = threadIdx.x;
  const int base = blockIdx.x * 64;

  // stage 1: stroke_out = relu(acc)
  for (int i = tid; i < 64 * 32; i += 128) {
    int node = base + (i >> 5);
    float v = (node < n) ? acc[node * 32 + (i & 31)] : 0.0f;
    so[i] = fmaxf(v, 0.0f);
  }
  __syncthreads();

  const int wave = tid >> 5, lane = tid & 31;
  const int half = lane >> 4, lrow = lane & 15;
  const int w16 = wave * 16;

  // stage 2: feats[16,64] = so[16,32] @ Wh[32,64] + bh
#pragma unroll
  for (int nt = 0; nt < 4; ++nt) {
    v8f c = {0, 0, 0, 0, 0, 0, 0, 0};
    const int col = nt * 16 + lrow;
#pragma unroll
    for (int k0 = 0; k0 < 32; k0 += 4) {
      const int ka = k0 + half * 2;
      v2f a = *(const v2f*)&so[(w16 + lrow) * 32 + ka];   // ds_load_b64
      v2f b; b.x = Wh[ka * 64 + col]; b.y = Wh[(ka + 1) * 64 + col];
      c = wmma4(a, b, c);
    }
    const float bb = bh[col];
#pragma unroll
    for (int r = 0; r < 8; ++r) fe[(w16 + r + half * 8) * 64 + col] = c[r] + bb;
  }
  __syncthreads();

  // stage 3: h[16,128] = relu(feats @ W1[64,128] + b1)
  for (int nt = 0; nt < 8; ++nt) {
    v8f c = {0, 0, 0, 0, 0, 0, 0, 0};
    const int col = nt * 16 + lrow;
#pragma unroll
    for (int k0 = 0; k0 < 64; k0 += 4) {
      const int ka = k0 + half * 2;
      v2f a = *(const v2f*)&fe[(w16 + lrow) * 64 + ka];   // ds_load_b64
      v2f b; b.x = W1[ka * 128 + col]; b.y = W1[(ka + 1) * 128 + col];
      c = wmma4(a, b, c);
    }
    const float bb = b1[col];
#pragma unroll
    for (int r = 0; r < 8; ++r)
      hh[(w16 + r + half * 8) * 128 + col] = fmaxf(c[r] + bb, 0.0f);
  }
  __syncthreads();

  // stage 4: out = sigmoid(h @ W2[128,1] + b2); 2 lanes per node
  const int node = tid >> 1, part = tid & 1;
  float sacc = 0.0f;
#pragma unroll 8
  for (int j = 0; j < 64; ++j)
    sacc += hh[node * 128 + part * 64 + j] * W2[part * 64 + j];
  sacc += __shfl_xor(sacc, 1);
  if (part == 0) {
    int gn = base + node;
    if (gn < n) out[gn] = 1.0f / (1.0f + __expf(-(sacc + b2[0])));
  }
}

// ------------------------------------------------------------------- launcher
extern "C" void kernel_launch(void* const* d_in, const int* in_sizes, int n_in,
                              void* d_out, int out_size, void* d_ws, size_t ws_size,
                              hipStream_t stream) {
  (void)n_in; (void)out_size; (void)ws_size;
  const float* x_stroke = (const float*)d_in[0];
  const float* x_brep   = (const float*)d_in[1];
  const float* sw       = (const float*)d_in[2];
  const float* W_int = (const float*)d_in[3];  const float* b_int = (const float*)d_in[4];
  const float* W_tp  = (const float*)d_in[5];  const float* b_tp  = (const float*)d_in[6];
  const float* W_rb  = (const float*)d_in[7];  const float* b_rb  = (const float*)d_in[8];
  // d_in[9..10] = W_bc/b_bc and d_in[25..26] = bc edges: dead in the reference
  // (brep_out is deleted and never reaches the decoder) -> skipped entirely.
  const float* W_sc  = (const float*)d_in[11]; const float* b_sc  = (const float*)d_in[12];
  const float* W_head= (const float*)d_in[13]; const float* b_head= (const float*)d_in[14];
  const float* W_d1  = (const float*)d_in[15]; const float* b_d1  = (const float*)d_in[16];
  const float* W_d2  = (const float*)d_in[17]; const float* b_d2  = (const float*)d_in[18];
  const int* ei_int_src = (const int*)d_in[19]; const int* ei_int_dst = (const int*)d_in[20];
  const int* ei_tp_src  = (const int*)d_in[21]; const int* ei_tp_dst  = (const int*)d_in[22];
  const int* ei_rb_src  = (const int*)d_in[23]; const int* ei_rb_dst  = (const int*)d_in[24];
  const int* ei_sc_src  = (const int*)d_in[27]; const int* ei_sc_dst  = (const int*)d_in[28];

  const int NS = in_sizes[2];          // stroke_weights is [N_S,1]
  const int NB = in_sizes[1] / 32;
  const int E_int = in_sizes[19], E_tp = in_sizes[21];
  const int E_rb  = in_sizes[23], E_sc = in_sizes[27];
  const int NS32 = NS * 32;

  // Workspace carve-out (sequential edge-type processing reuses t/s/c):
  float* xs  = (float*)d_ws;           // [NS,32] weighted strokes
  float* acc = xs + NS32;              // [NS,32] running xs + sum(h_*)
  float* t   = acc + NS32;             // [max(NS,NB),32] transformed src feats
  float* s   = t + NS32;               // [NS,32] per-type accumulator
  float* c   = s + NS32;               // [NS] edge counts for means
  float* outp = (float*)d_out;

  auto cdiv = [](int a, int b) { return (a + b - 1) / b; };
  const uint32_t NEG_INF = 0xFF800000u;

  k_xs<<<cdiv(NS32, 256), 256, 0, stream>>>(x_stroke, sw, xs, acc, NS32);

  // intersects: stroke->stroke, mean
  k_gemm32<<<cdiv(NS, 128), 256, 0, stream>>>(xs, W_int, b_int, t, NS);
  k_fill<<<cdiv(NS32, 256), 256, 0, stream>>>((uint32_t*)s, 0u, NS32);
  k_fill<<<cdiv(NS, 256), 256, 0, stream>>>((uint32_t*)c, 0u, NS);
  k_scatter_add<<<cdiv(E_int * 32, 256), 256, 0, stream>>>(t, ei_int_src, ei_int_dst, s, c, E_int);
  k_acc_mean<<<cdiv(NS32, 256), 256, 0, stream>>>(acc, s, c, NS32);

  // temp_previous: stroke->stroke, sum
  k_gemm32<<<cdiv(NS, 128), 256, 0, stream>>>(xs, W_tp, b_tp, t, NS);
  k_fill<<<cdiv(NS32, 256), 256, 0, stream>>>((uint32_t*)s, 0u, NS32);
  k_scatter_add<<<cdiv(E_tp * 32, 256), 256, 0, stream>>>(t, ei_tp_src, ei_tp_dst, s, nullptr, E_tp);
  k_acc_sum<<<cdiv(NS32, 256), 256, 0, stream>>>(acc, s, NS32);

  // represented_by: brep->stroke, mean (uses raw x_brep, not weighted)
  k_gemm32<<<cdiv(NB, 128), 256, 0, stream>>>(x_brep, W_rb, b_rb, t, NB);
  k_fill<<<cdiv(NS32, 256), 256, 0, stream>>>((uint32_t*)s, 0u, NS32);
  k_fill<<<cdiv(NS, 256), 256, 0, stream>>>((uint32_t*)c, 0u, NS);
  k_scatter_add<<<cdiv(E_rb * 32, 256), 256, 0, stream>>>(t, ei_rb_src, ei_rb_dst, s, c, E_rb);
  k_acc_mean<<<cdiv(NS32, 256), 256, 0, stream>>>(acc, s, c, NS32);

  // strokecoplanar: stroke->stroke, max (empty segments -> 0)
  k_gemm32<<<cdiv(NS, 128), 256, 0, stream>>>(xs, W_sc, b_sc, t, NS);
  k_fill<<<cdiv(NS32, 256), 256, 0, stream>>>((uint32_t*)s, NEG_INF, NS32);
  k_scatter_max<<<cdiv(E_sc * 32, 256), 256, 0, stream>>>(t, ei_sc_src, ei_sc_dst, s, E_sc);
  k_acc_max<<<cdiv(NS32, 256), 256, 0, stream>>>(acc, s, NS32);

  // fused relu-residual + head + MLP decoder + sigmoid
  k_decoder<<<cdiv(NS, 64), 128, 0, stream>>>(acc, W_head, b_head, W_d1, b_d1,
                                              W_d2, b_d2, outp, NS);
}